// GAT_L2_intervention_63831803953272
// MI455X (gfx1250) — compile-verified
//
#include <hip/hip_runtime.h>
#include <hip/hip_bf16.h>
#include <math.h>

// ---------------------------------------------------------------------------
// GAT 2-layer forward for MI455X (gfx1250, wave32).
// GEMMs: V_WMMA_F32_16X16X4_F32, W staged once per workgroup into LDS in
// pre-swizzled B-fragment layout (ds_load_b64 in the hot loop).
// Edge phase: f32 global atomics; all intermediates L2-resident (192 MB L2).
// ---------------------------------------------------------------------------

typedef __attribute__((ext_vector_type(2))) float v2f;
typedef __attribute__((ext_vector_type(8))) float v8f;

#define NEG_SLOPE 0.2f

// ---------------- f32 atomic max via monotonic int punning -----------------
__device__ __forceinline__ void atomicMaxFloat(float* addr, float value) {
    if (value >= 0.0f) {
        atomicMax((int*)addr, __float_as_int(value));
    } else {
        atomicMin((unsigned int*)addr, (unsigned int)__float_as_int(value));
    }
}

// ---------------- WMMA GEMM: C[M,NC] = A[M,K] @ B[K,NC] --------------------
// W is staged in LDS in WMMA B-fragment layout: for k-step ks, tile t, lane l
// the lane's (B[k0][n], B[k0+1][n]) pair is one contiguous 8-byte LDS entry.
// One wave computes a 16-row block and NC/16 column tiles per iteration.
template<int K, int NC>
__global__ void gemm_wmma_lds(const float* __restrict__ A,
                              const float* __restrict__ B,
                              float* __restrict__ C, int M) {
    constexpr int NT = NC / 16;
    constexpr int KS = K / 4;                    // k-steps of 4
    __shared__ v2f bfrag_lds[KS * NT * 32];      // 64 KB (NC=128) / 32 KB (NC=64)

    const int tid  = threadIdx.x;
    const int lane = tid & 31;
    const int half = lane >> 4;                  // 0: K pair {0,1}; 1: {2,3}
    const int lrow = lane & 15;                  // M (A) / N (B,C) in tile

    // ---- stage W -> LDS (fragment layout), once per workgroup ----
    for (int e = tid; e < KS * NT * 32; e += blockDim.x) {
        const int l    = e & 31;
        const int t    = (e >> 5) % NT;
        const int ks   = (e >> 5) / NT;
        const int krow = ks * 4 + (l >> 4) * 2;
        const int col  = t * 16 + (l & 15);
        v2f b;
        b.x = B[(size_t)krow * NC + col];
        b.y = B[(size_t)(krow + 1) * NC + col];
        bfrag_lds[e] = b;
    }
    __syncthreads();

    const int wave  = blockIdx.x * (blockDim.x >> 5) + (tid >> 5);
    const int nwave = gridDim.x * (blockDim.x >> 5);

    for (int rb = wave; rb < (M >> 4); rb += nwave) {
        const int row0 = rb * 16;
        v8f acc[NT];
#pragma unroll
        for (int t = 0; t < NT; ++t)
            acc[t] = (v8f){0.f,0.f,0.f,0.f,0.f,0.f,0.f,0.f};

        for (int ks = 0; ks < KS; ++ks) {
            // A fragment: 16x4 f32 (b64 global load, contiguous pair)
            const float* ap = A + (size_t)(row0 + lrow) * K + ks * 4 + half * 2;
            v2f afrag; afrag.x = ap[0]; afrag.y = ap[1];
            const v2f* bl = &bfrag_lds[(size_t)ks * NT * 32 + lane];
#pragma unroll
            for (int t = 0; t < NT; ++t) {
                v2f bfrag = bl[t * 32];          // ds_load_b64, immediate offset
                acc[t] = __builtin_amdgcn_wmma_f32_16x16x4_f32(
                    /*neg_a=*/false, afrag, /*neg_b=*/false, bfrag,
                    /*c_mod=*/(short)0, acc[t],
                    /*reuse_a=*/false, /*reuse_b=*/false);
            }
        }
        // Store D: VGPR i -> M = i + half*8, N = lrow
#pragma unroll
        for (int t = 0; t < NT; ++t) {
#pragma unroll
            for (int i = 0; i < 8; ++i) {
                C[(size_t)(row0 + i + half * 8) * NC + t * 16 + lrow] = acc[t][i];
            }
        }
    }
}

// ---------------- per-node attention logits --------------------------------
__global__ void node_alpha(const float* __restrict__ h,
                           const float* __restrict__ a_src,
                           const float* __restrict__ a_dst,
                           float* __restrict__ as, float* __restrict__ ad,
                           int Nn, int H, int C) {
    int i = blockIdx.x * blockDim.x + threadIdx.x;
    if (i >= Nn * H) return;
    int n = i / H, hh = i % H;
    const float* hp = h + (size_t)n * H * C + (size_t)hh * C;
    const float* s1 = a_src + hh * C;
    const float* s2 = a_dst + hh * C;
    float accS = 0.f, accD = 0.f;
    for (int c = 0; c < C; ++c) {
        float v = hp[c];
        accS += v * s1[c];
        accD += v * s2[c];
    }
    as[i] = accS;
    ad[i] = accD;
}

// ---------------- utility fills --------------------------------------------
__global__ void fill_f32(float* p, float v, long n) {
    long i = (long)blockIdx.x * blockDim.x + threadIdx.x;
    long stride = (long)gridDim.x * blockDim.x;
    for (; i < n; i += stride) p[i] = v;
}

__global__ void init_bias(float* p, const float* __restrict__ b, int Nn, int Cc) {
    long i = (long)blockIdx.x * blockDim.x + threadIdx.x;
    long n = (long)Nn * Cc;
    long stride = (long)gridDim.x * blockDim.x;
    for (; i < n; i += stride) p[i] = b[i % Cc];
}

__global__ void elu_inplace(float* p, long n) {
    long i = (long)blockIdx.x * blockDim.x + threadIdx.x;
    long stride = (long)gridDim.x * blockDim.x;
    for (; i < n; i += stride) {
        float x = p[i];
        p[i] = x > 0.f ? x : (expf(x) - 1.f);
    }
}

// ---------------- edge phase 1: leaky-relu logit + segment max -------------
__global__ void edge_alpha(const float* __restrict__ as, const float* __restrict__ ad,
                           const int* __restrict__ srcIdx, const int* __restrict__ dstIdx,
                           int E, int Nn, int H,
                           float* __restrict__ aE, float* __restrict__ amax) {
    int i = blockIdx.x * blockDim.x + threadIdx.x;
    int total = (E + Nn) * H;
    if (i >= total) return;
    int e = i / H, hh = i % H;
    int s, d;
    if (e < E) { s = srcIdx[e]; d = dstIdx[e]; } else { s = e - E; d = s; }
    float a = as[(size_t)s * H + hh] + ad[(size_t)d * H + hh];
    a = a > 0.f ? a : NEG_SLOPE * a;
    aE[i] = a;
    atomicMaxFloat(&amax[(size_t)d * H + hh], a);
}

// ---------------- edge phase 2: exp + segment sum --------------------------
__global__ void edge_exp(float* __restrict__ aE,
                         const float* __restrict__ amax,
                         const int* __restrict__ dstIdx,
                         int E, int Nn, int H,
                         float* __restrict__ den) {
    int i = blockIdx.x * blockDim.x + threadIdx.x;
    int total = (E + Nn) * H;
    if (i >= total) return;
    int e = i / H, hh = i % H;
    int d = (e < E) ? dstIdx[e] : (e - E);
    float ex = expf(aE[i] - amax[(size_t)d * H + hh]);
    aE[i] = ex;
    atomicAdd(&den[(size_t)d * H + hh], ex);
}

// ---------------- edge phase 3: normalize + weighted scatter-add -----------
// One wave per edge; each lane owns H*C/32 consecutive channels, so the
// h[src] gather vectorizes to b128/b64 and the atomic adds are coalesced.
template<int H, int C>
__global__ void edge_aggregate(const float* __restrict__ aE,
                               const float* __restrict__ den,
                               const int* __restrict__ srcIdx,
                               const int* __restrict__ dstIdx,
                               int E, int Nn,
                               const float* __restrict__ hsrc,
                               float* __restrict__ accum) {
    constexpr int HC = H * C;
    constexpr int V  = HC / 32;   // channels per lane
    const int lane = threadIdx.x & 31;
    const int w    = (blockIdx.x * blockDim.x + threadIdx.x) >> 5;
    const int gw   = (gridDim.x * blockDim.x) >> 5;
    const int Etot = E + Nn;
    const int c0   = lane * V;
    const int head = c0 / C;      // constant per lane for our shapes

    for (int e = w; e < Etot; e += gw) {
        // prefetch next iteration's index data (global_prefetch_b8)
        if (e + gw < E) {
            __builtin_prefetch(&srcIdx[e + gw], 0, 0);
            __builtin_prefetch(&dstIdx[e + gw], 0, 0);
        }
        int s, d;
        if (e < E) { s = srcIdx[e]; d = dstIdx[e]; } else { s = e - E; d = s; }
        float attn = aE[(size_t)e * H + head] / den[(size_t)d * H + head];
        const float* hp = hsrc + (size_t)s * HC + c0;
        float* op = accum + (size_t)d * HC + c0;
#pragma unroll
        for (int v = 0; v < V; ++v) atomicAdd(op + v, attn * hp[v]);
    }
}

// ---------------------------------------------------------------------------
extern "C" void kernel_launch(void* const* d_in, const int* in_sizes, int n_in,
                              void* d_out, int out_size, void* d_ws, size_t ws_size,
                              hipStream_t stream) {
    const float* x    = (const float*)d_in[0];
    const int*   ei   = (const int*)d_in[1];
    const float* W1   = (const float*)d_in[2];
    const float* aS1  = (const float*)d_in[3];
    const float* aD1  = (const float*)d_in[4];
    const float* b1   = (const float*)d_in[5];
    const float* W2   = (const float*)d_in[6];
    const float* aS2  = (const float*)d_in[7];
    const float* aD2  = (const float*)d_in[8];
    const float* b2   = (const float*)d_in[9];
    float* out = (float*)d_out;

    const int IN_C = 128, H1 = 4, HID = 32, HC1 = H1 * HID, OUT_C = 64;
    const int Nn   = in_sizes[0] / IN_C;       // 50000
    const int E    = in_sizes[1] / 2;          // 800000
    const int Etot = E + Nn;                   // self-loops appended
    const int* srcIdx = ei;
    const int* dstIdx = ei + E;

    // ---- workspace carve (floats) ----
    float* p    = (float*)d_ws;
    float* h1   = p; p += (size_t)Nn * HC1;    // layer-1 features
    float* acc1 = p; p += (size_t)Nn * HC1;    // layer-1 accum -> h2 after ELU
    float* g2   = p; p += (size_t)Nn * OUT_C;  // layer-2 features
    float* as1  = p; p += (size_t)Nn * H1;
    float* ad1  = p; p += (size_t)Nn * H1;
    float* am1  = p; p += (size_t)Nn * H1;
    float* dn1  = p; p += (size_t)Nn * H1;
    float* as2  = p; p += (size_t)Nn;
    float* ad2  = p; p += (size_t)Nn;
    float* am2  = p; p += (size_t)Nn;
    float* dn2  = p; p += (size_t)Nn;
    float* aE   = p; p += (size_t)Etot * H1;   // reused (H=1) for layer 2

    const int BLK = 256;
    auto cdiv = [](long a, long b) { return (int)((a + b - 1) / b); };

    const int gemmGrid = cdiv(Nn / 16, BLK / 32);        // 1 row-block per wave
    const int aggGrid  = 8192;                           // grid-stride over edges

    // ================= Layer 1 =================
    gemm_wmma_lds<128, 128><<<gemmGrid, BLK, 0, stream>>>(x, W1, h1, Nn);
    node_alpha<<<cdiv((long)Nn * H1, BLK), BLK, 0, stream>>>(h1, aS1, aD1, as1, ad1, Nn, H1, HID);
    fill_f32<<<cdiv((long)Nn * H1, BLK), BLK, 0, stream>>>(am1, -INFINITY, (long)Nn * H1);
    fill_f32<<<cdiv((long)Nn * H1, BLK), BLK, 0, stream>>>(dn1, 0.f, (long)Nn * H1);
    init_bias<<<cdiv((long)Nn * HC1, BLK), BLK, 0, stream>>>(acc1, b1, Nn, HC1);

    edge_alpha<<<cdiv((long)Etot * H1, BLK), BLK, 0, stream>>>(as1, ad1, srcIdx, dstIdx, E, Nn, H1, aE, am1);
    edge_exp<<<cdiv((long)Etot * H1, BLK), BLK, 0, stream>>>(aE, am1, dstIdx, E, Nn, H1, dn1);
    edge_aggregate<4, 32><<<aggGrid, BLK, 0, stream>>>(aE, dn1, srcIdx, dstIdx, E, Nn, h1, acc1);
    elu_inplace<<<cdiv((long)Nn * HC1, BLK), BLK, 0, stream>>>(acc1, (long)Nn * HC1);

    // ================= Layer 2 =================
    gemm_wmma_lds<128, 64><<<gemmGrid, BLK, 0, stream>>>(acc1, W2, g2, Nn);
    node_alpha<<<cdiv((long)Nn, BLK), BLK, 0, stream>>>(g2, aS2, aD2, as2, ad2, Nn, 1, OUT_C);
    fill_f32<<<cdiv((long)Nn, BLK), BLK, 0, stream>>>(am2, -INFINITY, (long)Nn);
    fill_f32<<<cdiv((long)Nn, BLK), BLK, 0, stream>>>(dn2, 0.f, (long)Nn);
    init_bias<<<cdiv((long)Nn * OUT_C, BLK), BLK, 0, stream>>>(out, b2, Nn, OUT_C);

    edge_alpha<<<cdiv((long)Etot, BLK), BLK, 0, stream>>>(as2, ad2, srcIdx, dstIdx, E, Nn, 1, aE, am2);
    edge_exp<<<cdiv((long)Etot, BLK), BLK, 0, stream>>>(aE, am2, dstIdx, E, Nn, 1, dn2);
    edge_aggregate<1, 64><<<aggGrid, BLK, 0, stream>>>(aE, dn2, srcIdx, dstIdx, E, Nn, g2, out);
}